// GCNNet_20581483283116
// MI455X (gfx1250) — compile-verified
//
#include <hip/hip_runtime.h>

typedef __attribute__((ext_vector_type(2))) float v2f;
typedef __attribute__((ext_vector_type(8))) float v8f;

// ---------------- degree / norm ----------------
__global__ void k_init_one(float* deg, int N) {
    int i = blockIdx.x * blockDim.x + threadIdx.x;
    if (i < N) deg[i] = 1.0f;   // self-loop weight
}

__global__ void k_deg_accum(const int* __restrict__ dst, float* deg, int E) {
    int e = blockIdx.x * blockDim.x + threadIdx.x;
    if (e < E) atomicAdd(&deg[dst[e]], 1.0f);
}

__global__ void k_rsqrt_inplace(float* deg, int N) {
    int i = blockIdx.x * blockDim.x + threadIdx.x;
    if (i < N) deg[i] = rsqrtf(deg[i]);
}

__global__ void k_edge_norm(const int* __restrict__ src, const int* __restrict__ dst,
                            const float* __restrict__ dis, float* __restrict__ nrm, int E) {
    int e = blockIdx.x * blockDim.x + threadIdx.x;
    if (e < E) nrm[e] = dis[src[e]] * dis[dst[e]];
}

// ---------------- input projection + PReLU + skip ----------------
__global__ void k_input_proj(const float* __restrict__ num_x, const float* __restrict__ num_mask,
                             const float* __restrict__ x, const float* __restrict__ Wn,
                             const float* __restrict__ bn, const float* __restrict__ a,
                             float* __restrict__ h, int total) {
    int i = blockIdx.x * blockDim.x + threadIdx.x;
    if (i >= total) return;
    int n = i >> 7, d = i & 127;
    float z = num_x[n] * num_mask[n] * Wn[d] + bn[d];
    z = (z >= 0.0f) ? z : a[d] * z;
    h[i] = z + x[i];
}

// ---------------- fp32 WMMA GEMM: t[N,128] = h[N,128] @ W[128,128] ----------------
// One wave computes a 16-row x 128-col strip: 8 column tiles, K=128 in steps of 4.
// W is staged in LDS K-pair-interleaved: element (k, c) -> Wl[(k>>1)*256 + 2c + (k&1)],
// so every WMMA B fragment ((K,K+1) x col) is one aligned ds_load_b64 into a VGPR pair.
__global__ __launch_bounds__(128) void k_gemm128_wmma(const float* __restrict__ h,
                                                      const float* __restrict__ W,
                                                      float* __restrict__ t,
                                                      int nRowTiles) {
    __shared__ float Wl[128 * 128];           // 64 KB of 320 KB WGP LDS
    {
        int tid = threadIdx.x;
        v2f* Wlv = (v2f*)Wl;
        #pragma unroll
        for (int i = 0; i < 64; ++i) {        // 8192 float2 pairs / 128 threads
            int idx = i * 128 + tid;
            int c = idx & 127;                // column
            int p = idx >> 7;                 // k-pair index (0..63)
            v2f pr;
            pr.x = W[(2 * p)     * 128 + c];  // coalesced across threads
            pr.y = W[(2 * p + 1) * 128 + c];
            Wlv[p * 128 + c] = pr;            // single ds_store_b64
        }
    }
    __syncthreads();

    const int wave = threadIdx.x >> 5;
    const int lane = threadIdx.x & 31;
    const int rowTile = blockIdx.x * 4 + wave;      // wave-uniform guard: EXEC all-1s
    if (rowTile >= nRowTiles) return;

    const int r15   = lane & 15;
    const int khalf = (lane >> 4) << 1;             // 0 or 2: A lanes 16-31 hold K=2,3
    const float* hrow = h + (size_t)(rowTile * 16 + r15) * 128;
    const v2f* Wlv = (const v2f*)Wl;

    v8f acc[8] = {};
    for (int k0 = 0; k0 < 128; k0 += 4) {
        // A fragment: (k0+khalf) is even, hrow 512B-aligned -> one global_load_b64
        v2f a = *(const v2f*)(hrow + k0 + khalf);
        const v2f* wrow = Wlv + (((k0 + khalf) >> 1) * 128 + r15);
        #pragma unroll
        for (int j = 0; j < 8; ++j) {               // 8 column tiles of 16
            v2f bfrag = wrow[j * 16];               // one ds_load_b64, pair-aligned
            acc[j] = __builtin_amdgcn_wmma_f32_16x16x4_f32(
                false, a, false, bfrag, (short)0, acc[j], false, false);
        }
    }

    // C/D layout: lane in [0,15]: (M=r, N=lane); lane in [16,31]: (M=r+8, N=lane-16)
    const int mhalf = (lane >> 4) << 3;
    float* trow = t + (size_t)(rowTile * 16 + mhalf) * 128 + r15;
    #pragma unroll
    for (int j = 0; j < 8; ++j)
        #pragma unroll
        for (int r = 0; r < 8; ++r)
            trow[(size_t)r * 128 + j * 16] = acc[j][r];
}

// ---------------- acc = self_norm * t + bias ----------------
__global__ void k_agg_init(const float* __restrict__ t, const float* __restrict__ dis,
                           const float* __restrict__ bias, float* __restrict__ acc, int total) {
    int i = blockIdx.x * blockDim.x + threadIdx.x;
    if (i >= total) return;
    int n = i >> 7, c = i & 127;
    float sn = dis[n] * dis[n];
    acc[i] = sn * t[i] + bias[c];
}

// ---------------- edge scatter: one wave per edge, 4 floats per lane ----------------
__global__ void k_edge_scatter_vec(const int* __restrict__ src, const int* __restrict__ dst,
                                   const float* __restrict__ nrm, const float* __restrict__ t,
                                   float* __restrict__ acc, int E) {
    int wid  = (blockIdx.x * blockDim.x + threadIdx.x) >> 5;
    int lane = threadIdx.x & 31;
    if (wid >= E) return;
    int s = src[wid], d = dst[wid];
    float w = nrm[wid];
    const float4 v = ((const float4*)(t + (size_t)s * 128))[lane];
    float* ap = acc + (size_t)d * 128 + lane * 4;
    atomicAdd(ap + 0, w * v.x);
    atomicAdd(ap + 1, w * v.y);
    atomicAdd(ap + 2, w * v.z);
    atomicAdd(ap + 3, w * v.w);
}

// ---------------- PReLU in place ----------------
__global__ void k_prelu(float* __restrict__ h, const float* __restrict__ a, int total) {
    int i = blockIdx.x * blockDim.x + threadIdx.x;
    if (i >= total) return;
    float v = h[i];
    h[i] = (v >= 0.0f) ? v : a[i & 127] * v;
}

// ---------------- layer 3: per-node dot with W3 ----------------
__global__ void k_dot_w3(const float* __restrict__ h, const float* __restrict__ W3,
                         float* __restrict__ t3, int N) {
    int wid  = (blockIdx.x * blockDim.x + threadIdx.x) >> 5;
    int lane = threadIdx.x & 31;
    if (wid >= N) return;
    const float4 hv = ((const float4*)(h + (size_t)wid * 128))[lane];
    const float4 wv = ((const float4*)W3)[lane];
    float s = hv.x * wv.x + hv.y * wv.y + hv.z * wv.z + hv.w * wv.w;
    #pragma unroll
    for (int off = 16; off > 0; off >>= 1) s += __shfl_xor(s, off, 32);
    if (lane == 0) t3[wid] = s;
}

__global__ void k_out_init(const float* __restrict__ t3, const float* __restrict__ dis,
                           const float* __restrict__ b3, float* __restrict__ out, int N) {
    int i = blockIdx.x * blockDim.x + threadIdx.x;
    if (i < N) out[i] = dis[i] * dis[i] * t3[i] + b3[0];
}

__global__ void k_edge_scatter_scalar(const int* __restrict__ src, const int* __restrict__ dst,
                                      const float* __restrict__ nrm, const float* __restrict__ t3,
                                      float* __restrict__ out, int E) {
    int e = blockIdx.x * blockDim.x + threadIdx.x;
    if (e < E) atomicAdd(&out[dst[e]], nrm[e] * t3[src[e]]);
}

// ---------------- launcher ----------------
extern "C" void kernel_launch(void* const* d_in, const int* in_sizes, int n_in,
                              void* d_out, int out_size, void* d_ws, size_t ws_size,
                              hipStream_t stream) {
    const float* num_x    = (const float*)d_in[0];
    const float* num_mask = (const float*)d_in[1];
    const float* x        = (const float*)d_in[2];
    const int*   edge     = (const int*)d_in[3];
    const float* W_num    = (const float*)d_in[4];
    const float* b_num    = (const float*)d_in[5];
    const float* a_in     = (const float*)d_in[6];
    const float* W1       = (const float*)d_in[7];
    const float* b1       = (const float*)d_in[8];
    const float* a1       = (const float*)d_in[9];
    const float* W2       = (const float*)d_in[10];
    const float* b2       = (const float*)d_in[11];
    const float* a2       = (const float*)d_in[12];
    const float* W3       = (const float*)d_in[13];
    const float* b3       = (const float*)d_in[14];

    const int N = in_sizes[1];          // num_mask is [N]
    const int E = in_sizes[3] / 2;      // edge_index is [2,E]
    const int* src = edge;
    const int* dst = edge + E;

    const int nRowTiles = (N + 15) / 16;
    const size_t padRows = (size_t)nRowTiles * 16;

    // workspace carve-up (all 256B aligned)
    auto al = [](size_t v) { return (v + 255) & ~(size_t)255; };
    char* p = (char*)d_ws;
    float* dis  = (float*)p; p += al((size_t)N * 4);
    float* nrm  = (float*)p; p += al((size_t)E * 4);
    float* hbuf = (float*)p; p += al(padRows * 128 * 4);
    float* tbuf = (float*)p; p += al(padRows * 128 * 4);
    float* t3   = (float*)p; p += al((size_t)N * 4);
    float* out  = (float*)d_out;

    const int TPB = 256;
    const int totND = N * 128;
    dim3 blk(TPB);
    auto gN  = dim3((N + TPB - 1) / TPB);
    auto gE  = dim3((E + TPB - 1) / TPB);
    auto gND = dim3((totND + TPB - 1) / TPB);
    auto gEw = dim3((int)(((size_t)E * 32 + TPB - 1) / TPB));   // wave per edge
    auto gNw = dim3((int)(((size_t)N * 32 + TPB - 1) / TPB));   // wave per node
    const int gemmBlocks = (nRowTiles + 3) / 4;                 // 4 waves / block

    // degrees + normalization
    k_init_one<<<gN, blk, 0, stream>>>(dis, N);
    k_deg_accum<<<gE, blk, 0, stream>>>(dst, dis, E);
    k_rsqrt_inplace<<<gN, blk, 0, stream>>>(dis, N);
    k_edge_norm<<<gE, blk, 0, stream>>>(src, dst, dis, nrm, E);

    // input projection
    k_input_proj<<<gND, blk, 0, stream>>>(num_x, num_mask, x, W_num, b_num, a_in, hbuf, totND);

    // layer 1
    k_gemm128_wmma<<<gemmBlocks, 128, 0, stream>>>(hbuf, W1, tbuf, nRowTiles);
    k_agg_init<<<gND, blk, 0, stream>>>(tbuf, dis, b1, hbuf, totND);
    k_edge_scatter_vec<<<gEw, blk, 0, stream>>>(src, dst, nrm, tbuf, hbuf, E);
    k_prelu<<<gND, blk, 0, stream>>>(hbuf, a1, totND);

    // layer 2
    k_gemm128_wmma<<<gemmBlocks, 128, 0, stream>>>(hbuf, W2, tbuf, nRowTiles);
    k_agg_init<<<gND, blk, 0, stream>>>(tbuf, dis, b2, hbuf, totND);
    k_edge_scatter_vec<<<gEw, blk, 0, stream>>>(src, dst, nrm, tbuf, hbuf, E);
    k_prelu<<<gND, blk, 0, stream>>>(hbuf, a2, totND);

    // layer 3 (H -> 1)
    k_dot_w3<<<gNw, blk, 0, stream>>>(hbuf, W3, t3, N);
    k_out_init<<<gN, blk, 0, stream>>>(t3, dis, b3, out, N);
    k_edge_scatter_scalar<<<gE, blk, 0, stream>>>(src, dst, nrm, t3, out, E);
}